// MHSA2D_62766652063799
// MI455X (gfx1250) — compile-verified
//
#include <hip/hip_runtime.h>

typedef __attribute__((ext_vector_type(2))) float v2f;
typedef __attribute__((ext_vector_type(8))) float v8f;

#define BATCH  8
#define CCH    384
#define NTOK   1024
#define NHEADS 8
#define HDIM   48
#define C3     (3 * CCH)

static __device__ __forceinline__ v8f wmma_f32(v2f a, v2f b, v8f c) {
  // D = A(16x4) * B(4x16) + C, full fp32
  return __builtin_amdgcn_wmma_f32_16x16x4_f32(false, a, false, b, (short)0, c,
                                               false, false);
}

// ---------------------------------------------------------------------------
// Stage 1: QKV projection.  qkv[o,n] = sum_c w_qkv[o,c] * x[b,c,n] + bias[o]
// Output scattered into q/k/v as [b][head][n][d] so attention loads along d
// are contiguous float2s.
// ---------------------------------------------------------------------------
__global__ void __launch_bounds__(128)
qkv_kernel(const float* __restrict__ x, const float* __restrict__ w,
           const float* __restrict__ bias, float* __restrict__ qo,
           float* __restrict__ ko, float* __restrict__ vo) {
  const int lane = threadIdx.x & 31;
  const int wv   = threadIdx.x >> 5;
  const int l15  = lane & 15;
  const int hi   = lane >> 4;
  const int n0   = blockIdx.x * 16;
  const int row0 = (blockIdx.y * 4 + wv) * 16;
  const int b    = blockIdx.z;
  const float* xb = x + (size_t)b * CCH * NTOK;

  v8f acc = {};
  for (int k = 0; k < CCH; k += 4) {
    const int ka = k + hi * 2;
    v2f a = *(const v2f*)(w + (size_t)(row0 + l15) * CCH + ka);
    v2f bf;
    bf.x = xb[(size_t)ka * NTOK + n0 + l15];
    bf.y = xb[(size_t)(ka + 1) * NTOK + n0 + l15];
    acc = wmma_f32(a, bf, acc);
  }

#pragma unroll
  for (int r = 0; r < 8; ++r) {
    const int o = row0 + r + 8 * hi;
    const float val = acc[r] + bias[o];
    const int three = o / CCH;
    const int rem   = o - three * CCH;
    const int head  = rem / HDIM;
    const int dd    = rem - head * HDIM;
    float* dst = (three == 0) ? qo : (three == 1) ? ko : vo;
    dst[(((size_t)b * NHEADS + head) * NTOK + (n0 + l15)) * HDIM + dd] = val;
  }
}

// ---------------------------------------------------------------------------
// Stage 2: flash attention. One wave handles 16 query rows; K/V streamed in
// 16-row tiles. Row reductions via shfl_xor within 16-lane halves (matches
// WMMA C layout row grouping). P transposed C->A layout via 1KB LDS tile.
// ---------------------------------------------------------------------------
__global__ void __launch_bounds__(32)
attn_kernel(const float* __restrict__ q, const float* __restrict__ k,
            const float* __restrict__ v, float* __restrict__ out) {
  __shared__ float pTile[16 * 16];
  const int lane = threadIdx.x & 31;
  const int l15  = lane & 15;
  const int hi   = lane >> 4;
  const int n0   = blockIdx.x * 16;
  const int h    = blockIdx.y;
  const int b    = blockIdx.z;
  const size_t base = ((size_t)b * NHEADS + h) * NTOK * HDIM;
  const float* Q = q + base;
  const float* K = k + base;
  const float* V = v + base;
  const float scale = 0.14433756729740643f;  // 48^-0.5

  // Preload Q fragments for all 12 K-steps (d = 48 = 12*4)
  v2f qa[12];
#pragma unroll
  for (int s = 0; s < 12; ++s) {
    const int ka = s * 4 + hi * 2;
    qa[s] = *(const v2f*)(Q + (size_t)(n0 + l15) * HDIM + ka);
  }

  float mrun[8], lrun[8];
#pragma unroll
  for (int r = 0; r < 8; ++r) { mrun[r] = -1e30f; lrun[r] = 0.0f; }
  v8f o0 = {}, o1 = {}, o2 = {};

  for (int m0 = 0; m0 < NTOK; m0 += 16) {
    // S = Q * K^T  (B fragment: lane m holds K[m0+m, ka..ka+1])
    v8f s = {};
#pragma unroll
    for (int st = 0; st < 12; ++st) {
      const int ka = st * 4 + hi * 2;
      v2f bf = *(const v2f*)(K + (size_t)(m0 + l15) * HDIM + ka);
      s = wmma_f32(qa[st], bf, s);
    }

    // Online softmax update per row (rows r+8*hi live in 16-lane halves)
    float fac[8];
#pragma unroll
    for (int r = 0; r < 8; ++r) {
      float xv = s[r] * scale;
      float mx = xv;
      mx = fmaxf(mx, __shfl_xor(mx, 1));
      mx = fmaxf(mx, __shfl_xor(mx, 2));
      mx = fmaxf(mx, __shfl_xor(mx, 4));
      mx = fmaxf(mx, __shfl_xor(mx, 8));
      const float mnew = fmaxf(mrun[r], mx);
      fac[r] = __expf(mrun[r] - mnew);
      mrun[r] = mnew;
      const float p = __expf(xv - mnew);
      float sum = p;
      sum += __shfl_xor(sum, 1);
      sum += __shfl_xor(sum, 2);
      sum += __shfl_xor(sum, 4);
      sum += __shfl_xor(sum, 8);
      lrun[r] = lrun[r] * fac[r] + sum;
      s[r] = p;
    }
#pragma unroll
    for (int r = 0; r < 8; ++r) {
      o0[r] *= fac[r]; o1[r] *= fac[r]; o2[r] *= fac[r];
    }

    // Transpose P from C-layout to A-layout through LDS
    __syncthreads();
#pragma unroll
    for (int r = 0; r < 8; ++r) pTile[(r + 8 * hi) * 16 + l15] = s[r];
    __syncthreads();

    // O += P * V  (3 tiles cover d = 48)
#pragma unroll
    for (int st = 0; st < 4; ++st) {
      const int ka = st * 4 + hi * 2;
      v2f pa = *(const v2f*)(&pTile[l15 * 16 + ka]);
      const float* vr0 = V + (size_t)(m0 + ka) * HDIM;
      const float* vr1 = V + (size_t)(m0 + ka + 1) * HDIM;
      v2f b0, b1, b2;
      b0.x = vr0[l15];      b0.y = vr1[l15];
      b1.x = vr0[16 + l15]; b1.y = vr1[16 + l15];
      b2.x = vr0[32 + l15]; b2.y = vr1[32 + l15];
      o0 = wmma_f32(pa, b0, o0);
      o1 = wmma_f32(pa, b1, o1);
      o2 = wmma_f32(pa, b2, o2);
    }
  }

  // Normalize and store as [b][c][n] for the projection GEMM
#pragma unroll
  for (int r = 0; r < 8; ++r) {
    const float inv = 1.0f / lrun[r];
    const int nn = n0 + r + 8 * hi;
    out[((size_t)b * CCH + (h * HDIM + 0  + l15)) * NTOK + nn] = o0[r] * inv;
    out[((size_t)b * CCH + (h * HDIM + 16 + l15)) * NTOK + nn] = o1[r] * inv;
    out[((size_t)b * CCH + (h * HDIM + 32 + l15)) * NTOK + nn] = o2[r] * inv;
  }
}

// ---------------------------------------------------------------------------
// Stage 3: output projection. out[o,n] = sum_c w_proj[o,c]*attn[c,n] + bias[o]
// ---------------------------------------------------------------------------
__global__ void __launch_bounds__(128)
proj_kernel(const float* __restrict__ a_in, const float* __restrict__ w,
            const float* __restrict__ bias, float* __restrict__ out) {
  const int lane = threadIdx.x & 31;
  const int wv   = threadIdx.x >> 5;
  const int l15  = lane & 15;
  const int hi   = lane >> 4;
  const int n0   = blockIdx.x * 16;
  const int row0 = (blockIdx.y * 4 + wv) * 16;
  const int b    = blockIdx.z;
  const float* ab = a_in + (size_t)b * CCH * NTOK;

  v8f acc = {};
  for (int k = 0; k < CCH; k += 4) {
    const int ka = k + hi * 2;
    v2f a = *(const v2f*)(w + (size_t)(row0 + l15) * CCH + ka);
    v2f bf;
    bf.x = ab[(size_t)ka * NTOK + n0 + l15];
    bf.y = ab[(size_t)(ka + 1) * NTOK + n0 + l15];
    acc = wmma_f32(a, bf, acc);
  }
#pragma unroll
  for (int r = 0; r < 8; ++r) {
    const int o = row0 + r + 8 * hi;
    out[((size_t)b * CCH + o) * NTOK + n0 + l15] = acc[r] + bias[o];
  }
}

// ---------------------------------------------------------------------------
extern "C" void kernel_launch(void* const* d_in, const int* in_sizes, int n_in,
                              void* d_out, int out_size, void* d_ws,
                              size_t ws_size, hipStream_t stream) {
  (void)in_sizes; (void)n_in; (void)out_size; (void)ws_size;
  const float* x      = (const float*)d_in[0];
  const float* w_qkv  = (const float*)d_in[1];
  const float* b_qkv  = (const float*)d_in[2];
  const float* w_proj = (const float*)d_in[3];
  const float* b_proj = (const float*)d_in[4];
  float* out = (float*)d_out;

  const size_t seg = (size_t)BATCH * NHEADS * NTOK * HDIM;  // == B*C*N floats
  float* qbuf = (float*)d_ws;
  float* kbuf = qbuf + seg;
  float* vbuf = kbuf + seg;
  float* abuf = vbuf + seg;

  dim3 g1(NTOK / 16, C3 / 64, BATCH);   // (64, 18, 8), 128 thr = 4 waves
  qkv_kernel<<<g1, 128, 0, stream>>>(x, w_qkv, b_qkv, qbuf, kbuf, vbuf);

  dim3 g2(NTOK / 16, NHEADS, BATCH);    // (64, 8, 8), 1 wave each
  attn_kernel<<<g2, 32, 0, stream>>>(qbuf, kbuf, vbuf, abuf);

  dim3 g3(NTOK / 16, CCH / 64, BATCH);  // (64, 6, 8)
  proj_kernel<<<g3, 128, 0, stream>>>(abuf, w_proj, b_proj, out);
}